// SubstitutionHead_34239479284417
// MI455X (gfx1250) — compile-verified
//
#include <hip/hip_runtime.h>
#include <math.h>
#include <stdint.h>

#define HEADD   64
#define T2D     8192
#define LEN1D   65536      // T2*8
#define MIXD    16384      // LEN1/4
#define LEN0D   131072     // MIX*8
#define NSEGD   1024       // LEN1/64

typedef float v2f __attribute__((ext_vector_type(2)));
typedef float v8f __attribute__((ext_vector_type(8)));

__device__ __forceinline__ float gelu_f(float x) {
  return 0.5f * x * (1.0f + erff(x * 0.70710678118654752440f));
}

// CDNA5 async global->LDS copy (ASYNCcnt-tracked), ISA §10.4 / §15.18.3.
__device__ __forceinline__ void async_ld_lds_b128(void* lds_ptr, const float* gptr) {
  unsigned lds = (unsigned)(uintptr_t)lds_ptr;   // low 32 bits of generic LDS addr = LDS offset
  asm volatile("global_load_async_to_lds_b128 %0, %1, off" :: "v"(lds), "v"(gptr) : "memory");
}
__device__ __forceinline__ void wait_asynccnt0() {
  asm volatile("s_wait_asynccnt 0x0" ::: "memory");
}

// ---------------- weight-prep kernels (tiny, run once per launch) ----------------

// Masked block-conv weight: out[(i*64+c)*512 + (j*64+o)] = W[j,o,c,i] * (i<=j)
__global__ void prep_blockconv_W(const float* __restrict__ W, float* __restrict__ out) {
  int t = blockIdx.x * blockDim.x + threadIdx.x;     // 0..262143
  int row = t >> 9;          // i*64+c
  int col = t & 511;         // j*64+o
  int i = row >> 6, c = row & 63;
  int j = col >> 6, o = col & 63;
  out[t] = (i <= j) ? W[(((size_t)((j << 6) + o) << 6) + c) * 8 + i] : 0.0f;
}

// Deconv weight: out[c*512 + (j*64+o)] = W[c,o,j]
__global__ void prep_deconv_W(const float* __restrict__ W, float* __restrict__ out) {
  int t = blockIdx.x * blockDim.x + threadIdx.x;
  int c = t >> 9;
  int col = t & 511;
  int j = col >> 6, o = col & 63;
  out[t] = W[((size_t)((c << 6) + o)) * 8 + j];
}

// Conv (Wdc) weight: out[(i*64+c)*64 + o] = Wdc[o,c,i]
__global__ void prep_conv_W(const float* __restrict__ W, float* __restrict__ out) {
  int t = blockIdx.x * blockDim.x + threadIdx.x;     // 0..32767
  int row = t >> 6;
  int o = t & 63;
  int i = row >> 6, c = row & 63;
  out[t] = W[((size_t)((o << 6) + c)) * 8 + i];
}

// 64x64 transpose: out[c*64+o] = W[o*64+c]
__global__ void prep_T64(const float* __restrict__ W, float* __restrict__ out) {
  int t = blockIdx.x * blockDim.x + threadIdx.x;     // 0..4095
  int c = t >> 6, o = t & 63;
  out[t] = W[(o << 6) + c];
}

// bias tile: out[j*64+o] = b[o] (512)
__global__ void prep_bias_tile(const float* __restrict__ b, float* __restrict__ out) {
  int t = blockIdx.x * blockDim.x + threadIdx.x;
  if (t < 512) out[t] = b[t & 63];
}

// ---------------- idx compaction: nonzero(val1==2, size=MIX, fill 0) ----------------

__global__ void k_seg_count(const int* __restrict__ val1, int* __restrict__ segcnt) {
  int s = blockIdx.x * blockDim.x + threadIdx.x;
  if (s >= NSEGD) return;
  int cnt = 0;
  const int* p = val1 + s * 64;
  for (int j = 0; j < 64; ++j) cnt += (p[j] == 2);
  segcnt[s] = cnt;
}

__global__ void k_seg_scan(const int* __restrict__ segcnt, int* __restrict__ segoff) {
  __shared__ int sh[NSEGD];
  int t = threadIdx.x;
  int own = segcnt[t];
  sh[t] = own;
  __syncthreads();
  for (int d = 1; d < NSEGD; d <<= 1) {
    int v = (t >= d) ? sh[t - d] : 0;
    __syncthreads();
    sh[t] += v;
    __syncthreads();
  }
  segoff[t] = sh[t] - own;     // exclusive prefix
}

__global__ void k_seg_write(const int* __restrict__ val1, const int* __restrict__ segoff,
                            int* __restrict__ idx) {
  int s = blockIdx.x * blockDim.x + threadIdx.x;
  if (s >= NSEGD) return;
  int off = segoff[s];
  const int* p = val1 + s * 64;
  for (int j = 0; j < 64; ++j) {
    if (p[j] == 2) {
      if (off < MIXD) idx[off] = s * 64 + j;
      ++off;
    }
  }
}

// ---------------- gathers / scatters ----------------

__global__ void k_gather_emb0(const int* __restrict__ val0, const float* __restrict__ tab,
                              float* __restrict__ emb0) {
  int t = blockIdx.x * blockDim.x + threadIdx.x;   // LEN0*64
  int r = t >> 6, c = t & 63;
  emb0[t] = tab[(val0[r] << 6) + c];
}

__global__ void k_scatter_down(const float* __restrict__ down, const int* __restrict__ idx,
                               float* __restrict__ emb1) {
  int t = blockIdx.x * blockDim.x + threadIdx.x;   // MIX*64
  int k = t >> 6, c = t & 63;
  emb1[((size_t)idx[k] << 6) + c] = down[t];
}

__global__ void k_gather_x0(const float* __restrict__ y1, const int* __restrict__ idx,
                            float* __restrict__ x0) {
  int t = blockIdx.x * blockDim.x + threadIdx.x;   // MIX*64
  int k = t >> 6, c = t & 63;
  x0[t] = y1[((size_t)idx[k] << 6) + c];
}

// ---------------- generic f32 WMMA GEMM (double-buffered, async B staging) ----------------
// C(MxN) = [gelu?](A) (MxK) * B (KxN) + bias[n] (+ add[m,n])
// Block = 256 threads (8 waves), 64x64 tile; wave w: ni=w&3, mi in {w>>2,(w>>2)+2}.
// K in chunks of 16; B tile staged global->LDS via GLOBAL_LOAD_ASYNC_TO_LDS_B128,
// A tile through VGPRs (GELU applied in flight); one barrier per chunk.
__global__ __launch_bounds__(256)
void gemm_wmma_f32(const float* __restrict__ A, const float* __restrict__ B,
                   const float* __restrict__ bias, const float* __restrict__ add,
                   float* __restrict__ C, int M, int N, int K, int gelu_in)
{
  __shared__ float As[2][64][16];
  __shared__ float Bs[2][16][68];
  (void)M;
  const int tid  = threadIdx.x;
  const int wave = tid >> 5;
  const int lane = tid & 31;
  const int half = lane >> 4;
  const int l15  = lane & 15;
  const int tileM = blockIdx.x * 64;
  const int tileN = blockIdx.y * 64;
  const int ni = wave & 3;
  const int mi = wave >> 2;       // 0 or 1; also handles mi+2

  // per-thread staging coordinates
  const int ar = tid >> 2;             // A row 0..63
  const int ak = (tid & 3) << 2;       // A col 0,4,8,12
  const int br = tid >> 4;             // B row 0..15
  const int bc = (tid & 15) << 2;      // B col 0..60

  const float* Aptr = A + (size_t)(tileM + ar) * K + ak;
  const float* Bptr = B + (size_t)br * N + tileN + bc;

  v8f acc0 = (v8f){0,0,0,0,0,0,0,0};
  v8f acc1 = (v8f){0,0,0,0,0,0,0,0};

  const int niter = K >> 4;

  // prologue: stage chunk 0
  {
    float4 v = *(const float4*)Aptr;
    if (gelu_in) { v.x = gelu_f(v.x); v.y = gelu_f(v.y); v.z = gelu_f(v.z); v.w = gelu_f(v.w); }
    As[0][ar][ak + 0] = v.x; As[0][ar][ak + 1] = v.y;
    As[0][ar][ak + 2] = v.z; As[0][ar][ak + 3] = v.w;
    async_ld_lds_b128(&Bs[0][br][bc], Bptr);
    wait_asynccnt0();
  }
  __syncthreads();

  for (int it = 0; it < niter; ++it) {
    const int cur = it & 1;
    const int nxt = cur ^ 1;
    const bool more = (it + 1) < niter;
    float4 av;
    if (more) {
      // issue next chunk's loads before the WMMA burst
      av = *(const float4*)(Aptr + (size_t)(it + 1) * 16);
      if (gelu_in) { av.x = gelu_f(av.x); av.y = gelu_f(av.y); av.z = gelu_f(av.z); av.w = gelu_f(av.w); }
      async_ld_lds_b128(&Bs[nxt][br][bc], Bptr + (size_t)(it + 1) * 16 * N);
    }
#pragma unroll
    for (int kk = 0; kk < 16; kk += 4) {
      const int ks = kk + 2 * half;   // VGPR0: K=ks, VGPR1: K=ks+1 (f32 WMMA lane layout)
      v2f a0 = { As[cur][(mi << 4) + l15][ks],       As[cur][(mi << 4) + l15][ks + 1] };
      v2f a1 = { As[cur][((mi + 2) << 4) + l15][ks], As[cur][((mi + 2) << 4) + l15][ks + 1] };
      v2f b  = { Bs[cur][ks][(ni << 4) + l15],       Bs[cur][ks + 1][(ni << 4) + l15] };
      acc0 = __builtin_amdgcn_wmma_f32_16x16x4_f32(false, a0, false, b, (short)0, acc0, false, false);
      acc1 = __builtin_amdgcn_wmma_f32_16x16x4_f32(false, a1, false, b, (short)0, acc1, false, false);
    }
    if (more) {
      As[nxt][ar][ak + 0] = av.x; As[nxt][ar][ak + 1] = av.y;
      As[nxt][ar][ak + 2] = av.z; As[nxt][ar][ak + 3] = av.w;
      wait_asynccnt0();
    }
    __syncthreads();
  }

  const int n = tileN + (ni << 4) + l15;
  const float bn = bias ? bias[n] : 0.0f;
#pragma unroll
  for (int v = 0; v < 8; ++v) {
    int m0 = tileM + (mi << 4) + v + 8 * half;
    int m1 = tileM + ((mi + 2) << 4) + v + 8 * half;
    size_t o0 = (size_t)m0 * N + n;
    size_t o1 = (size_t)m1 * N + n;
    float r0 = acc0[v] + bn;
    float r1 = acc1[v] + bn;
    if (add) { r0 += add[o0]; r1 += add[o1]; }
    C[o0] = r0;
    C[o1] = r1;
  }
}

// ---------------- fused head: out = gelu(gelu(y0)@Wl1T + bl1) @ Wl2^T + bl2 ----------------
__global__ __launch_bounds__(256)
void final_head(const float* __restrict__ y0, const float* __restrict__ Wl1T,
                const float* __restrict__ bl1, const float* __restrict__ Wl2,
                const float* __restrict__ bl2, float* __restrict__ out)
{
  __shared__ float Bs[64][65];
  __shared__ float Hs[8][16][65];
  const int tid  = threadIdx.x;
  const int wave = tid >> 5;
  const int lane = tid & 31;
  const int half = lane >> 4;
  const int l15  = lane & 15;
  const int base = blockIdx.x * 128 + wave * 16;

  for (int e = tid; e < 4096; e += 256) Bs[e >> 6][e & 63] = Wl1T[e];
  for (int e = lane; e < 1024; e += 32) {
    int r = e >> 6, c = e & 63;
    Hs[wave][r][c] = gelu_f(y0[(size_t)(base + r) * 64 + c]);
  }
  __syncthreads();

  v8f acc[4];
#pragma unroll
  for (int q = 0; q < 4; ++q) acc[q] = (v8f){0,0,0,0,0,0,0,0};
#pragma unroll
  for (int kk = 0; kk < 64; kk += 4) {
    const int ks = kk + 2 * half;
    v2f a = { Hs[wave][l15][ks], Hs[wave][l15][ks + 1] };
#pragma unroll
    for (int q = 0; q < 4; ++q) {
      v2f b = { Bs[ks][(q << 4) + l15], Bs[ks + 1][(q << 4) + l15] };
      acc[q] = __builtin_amdgcn_wmma_f32_16x16x4_f32(false, a, false, b, (short)0, acc[q], false, false);
    }
  }
  __syncthreads();
#pragma unroll
  for (int q = 0; q < 4; ++q) {
    int n = (q << 4) + l15;
    float bn = bl1[n];
#pragma unroll
    for (int v = 0; v < 8; ++v)
      Hs[wave][v + 8 * half][n] = gelu_f(acc[q][v] + bn);
  }
  __syncthreads();
  for (int p = lane; p < 48; p += 32) {
    int r = p / 3, t = p - 3 * r;
    float s = bl2[t];
    for (int c = 0; c < 64; ++c) s += Hs[wave][r][c] * Wl2[t * 64 + c];
    out[(size_t)(base + r) * 3 + t] = s;
  }
}

// ---------------- launch ----------------

extern "C" void kernel_launch(void* const* d_in, const int* in_sizes, int n_in,
                              void* d_out, int out_size, void* d_ws, size_t ws_size,
                              hipStream_t stream)
{
  (void)in_sizes; (void)n_in; (void)out_size; (void)ws_size;
  const float* x       = (const float*)d_in[0];
  const int*   value   = (const int*)d_in[1];
  const float* emb_tab = (const float*)d_in[4];
  const float* Wdc     = (const float*)d_in[5];
  const float* bdc     = (const float*)d_in[6];
  const float* bc0a_W  = (const float*)d_in[7];
  const float* bc0a_b  = (const float*)d_in[8];    // (8,64) contiguous == 512 bias
  const float* bc0b_W  = (const float*)d_in[9];
  const float* bc0b_b  = (const float*)d_in[10];
  const float* bc1a_W  = (const float*)d_in[11];
  const float* bc1a_b  = (const float*)d_in[12];
  const float* bc1b_W  = (const float*)d_in[13];
  const float* bc1b_b  = (const float*)d_in[14];
  const float* Wd1     = (const float*)d_in[15];
  const float* bd1     = (const float*)d_in[16];
  const float* Wp1     = (const float*)d_in[17];
  const float* bp1     = (const float*)d_in[18];
  const float* Wd0     = (const float*)d_in[19];
  const float* bd0     = (const float*)d_in[20];
  const float* Wp0     = (const float*)d_in[21];
  const float* bp0     = (const float*)d_in[22];
  const float* Wl1     = (const float*)d_in[23];
  const float* bl1     = (const float*)d_in[24];
  const float* Wl2     = (const float*)d_in[25];
  const float* bl2     = (const float*)d_in[26];
  float* out = (float*)d_out;

  // workspace carve-out (floats)
  float* ws = (float*)d_ws;
  size_t off = 0;
  float* f_emb0 = ws + off; off += (size_t)LEN0D * 64;  // emb0, later e0
  float* f_buf2 = ws + off; off += (size_t)LEN0D * 64;  // e0a, later y0pre/y0
  float* f_emb1 = ws + off; off += (size_t)LEN1D * 64;  // emb1, later e1
  float* f_buf4 = ws + off; off += (size_t)LEN1D * 64;  // e1a, later y1pre/y1
  float* f_down = ws + off; off += (size_t)MIXD * 64;   // down, later x0
  float* f_B0a  = ws + off; off += 512 * 512;
  float* f_B0b  = ws + off; off += 512 * 512;
  float* f_B1a  = ws + off; off += 512 * 512;
  float* f_B1b  = ws + off; off += 512 * 512;
  float* f_Bd1  = ws + off; off += 512 * 512;
  float* f_Bd0  = ws + off; off += 64 * 512;
  float* f_Bdc  = ws + off; off += 512 * 64;
  float* f_Wp1T = ws + off; off += 64 * 64;
  float* f_Wp0T = ws + off; off += 64 * 64;
  float* f_Wl1T = ws + off; off += 64 * 64;
  float* f_bd1t = ws + off; off += 512;
  float* f_bd0t = ws + off; off += 512;
  int* i_idx  = (int*)(ws + off);
  int* i_segc = i_idx + MIXD;
  int* i_sego = i_segc + NSEGD;

  const int BT = 256;

  // weight transforms
  prep_blockconv_W<<<262144 / BT, BT, 0, stream>>>(bc0a_W, f_B0a);
  prep_blockconv_W<<<262144 / BT, BT, 0, stream>>>(bc0b_W, f_B0b);
  prep_blockconv_W<<<262144 / BT, BT, 0, stream>>>(bc1a_W, f_B1a);
  prep_blockconv_W<<<262144 / BT, BT, 0, stream>>>(bc1b_W, f_B1b);
  prep_deconv_W<<<262144 / BT, BT, 0, stream>>>(Wd1, f_Bd1);
  prep_deconv_W<<<32768 / BT, BT, 0, stream>>>(Wd0, f_Bd0);
  prep_conv_W<<<32768 / BT, BT, 0, stream>>>(Wdc, f_Bdc);
  prep_T64<<<4096 / BT, BT, 0, stream>>>(Wp1, f_Wp1T);
  prep_T64<<<4096 / BT, BT, 0, stream>>>(Wp0, f_Wp0T);
  prep_T64<<<4096 / BT, BT, 0, stream>>>(Wl1, f_Wl1T);
  prep_bias_tile<<<2, BT, 0, stream>>>(bd1, f_bd1t);
  prep_bias_tile<<<2, BT, 0, stream>>>(bd0, f_bd0t);

  // idx = nonzero(val1 == 2, size=MIX, fill 0)
  k_seg_count<<<NSEGD / BT, BT, 0, stream>>>(value, i_segc);
  k_seg_scan<<<1, NSEGD, 0, stream>>>(i_segc, i_sego);
  hipMemsetAsync(i_idx, 0, MIXD * sizeof(int), stream);
  k_seg_write<<<NSEGD / BT, BT, 0, stream>>>(value, i_sego, i_idx);

  // emb0 gather, down conv (GEMM), emb1 scatter
  k_gather_emb0<<<(LEN0D * 64) / BT, BT, 0, stream>>>(value + LEN1D, emb_tab, f_emb0);
  gemm_wmma_f32<<<dim3(MIXD / 64, 1), BT, 0, stream>>>(f_emb0, f_Bdc, bdc, nullptr, f_down,
                                                       MIXD, 64, 512, 0);
  hipMemsetAsync(f_emb1, 0, (size_t)LEN1D * 64 * sizeof(float), stream);
  k_scatter_down<<<(MIXD * 64) / BT, BT, 0, stream>>>(f_down, i_idx, f_emb1);

  // block-convs as GEMMs
  gemm_wmma_f32<<<dim3(MIXD / 64, 8), BT, 0, stream>>>(f_emb0, f_B0a, bc0a_b, nullptr, f_buf2,
                                                       MIXD, 512, 512, 0);        // e0a
  gemm_wmma_f32<<<dim3(MIXD / 64, 8), BT, 0, stream>>>(f_buf2, f_B0b, bc0b_b, nullptr, f_emb0,
                                                       MIXD, 512, 512, 1);        // e0 (reuse emb0 buf)
  gemm_wmma_f32<<<dim3(T2D / 64, 8), BT, 0, stream>>>(f_emb1, f_B1a, bc1a_b, nullptr, f_buf4,
                                                      T2D, 512, 512, 1);          // e1a
  gemm_wmma_f32<<<dim3(T2D / 64, 8), BT, 0, stream>>>(f_buf4, f_B1b, bc1b_b, nullptr, f_emb1,
                                                      T2D, 512, 512, 1);          // e1 (reuse emb1 buf)

  // y1 = pw(gelu(deconv(gelu(x)))) + e1
  gemm_wmma_f32<<<dim3(T2D / 64, 8), BT, 0, stream>>>(x, f_Bd1, f_bd1t, nullptr, f_buf4,
                                                      T2D, 512, 512, 1);          // y1pre
  gemm_wmma_f32<<<dim3(LEN1D / 64, 1), BT, 0, stream>>>(f_buf4, f_Wp1T, bp1, f_emb1, f_buf4,
                                                        LEN1D, 64, 64, 1);        // y1 in-place

  // x0 = y1[idx]; y0 = pw(gelu(deconv(gelu(x0)))) + e0
  k_gather_x0<<<(MIXD * 64) / BT, BT, 0, stream>>>(f_buf4, i_idx, f_down);
  gemm_wmma_f32<<<dim3(MIXD / 64, 8), BT, 0, stream>>>(f_down, f_Bd0, f_bd0t, nullptr, f_buf2,
                                                       MIXD, 512, 64, 1);         // y0pre
  gemm_wmma_f32<<<dim3(LEN0D / 64, 1), BT, 0, stream>>>(f_buf2, f_Wp0T, bp0, f_emb0, f_buf2,
                                                        LEN0D, 64, 64, 1);        // y0 in-place

  // final fused head -> out (LEN0 x 3)
  final_head<<<LEN0D / 128, BT, 0, stream>>>(f_buf2, f_Wl1T, bl1, Wl2, bl2, out);
}